// ChannelConformer_9663676416726
// MI455X (gfx1250) — compile-verified
//
#include <hip/hip_runtime.h>
#include <math.h>

// Problem constants (from reference)
#define C_DIM   256
#define FF_DIM  128
#define B_DIM   4
#define SPATIAL 65536          // D*H*W = 16*64*64
#define EPS_LN  1e-5f

typedef __attribute__((ext_vector_type(2))) float v2f;
typedef __attribute__((ext_vector_type(8))) float v8f;

__device__ __forceinline__ float gelu_exact(float x) {
  return 0.5f * x * (1.0f + erff(x * 0.70710678118654752f));
}

// wave32 reduction of two running sums; result broadcast to all lanes
__device__ __forceinline__ void wave_reduce2(float& s, float& s2) {
  for (int off = 16; off > 0; off >>= 1) {
    s  += __shfl_down(s, off, 32);
    s2 += __shfl_down(s2, off, 32);
  }
  s  = __shfl(s, 0, 32);
  s2 = __shfl(s2, 0, 32);
}

// ---------------------------------------------------------------------------
// Kernel 1: tokens[b][c] = mean over 65536 spatial elements (streaming, b128)
// One block per (b,c) pair; contiguous 256KB per block.
// ---------------------------------------------------------------------------
__global__ __launch_bounds__(256) void tokens_mean_kernel(
    const float* __restrict__ x, float* __restrict__ tokens) {
  __shared__ float red[256];
  const int bc = blockIdx.x;                              // 0..1023
  const float4* p4 = (const float4*)(x + (size_t)bc * SPATIAL);
  float s = 0.0f;
  for (int i = threadIdx.x; i < SPATIAL / 4; i += 256) {  // 64 b128 loads/thread
    float4 v = p4[i];
    s += v.x + v.y + v.z + v.w;
  }
  red[threadIdx.x] = s;
  __syncthreads();
  for (int w = 128; w > 0; w >>= 1) {
    if (threadIdx.x < w) red[threadIdx.x] += red[threadIdx.x + w];
    __syncthreads();
  }
  if (threadIdx.x == 0) tokens[bc] = red[0] * (1.0f / (float)SPATIAL);
}

// ---------------------------------------------------------------------------
// fp32 WMMA tile: Out[0:16, n0:n0+16] = A(16xK) @ W^T + bias (+Res) (gelu?)
// A in LDS row-major [16][K]; W global row-major [N][K]; one wave per tile.
// Uses V_WMMA_F32_16X16X4_F32 (exact fp32).
// ---------------------------------------------------------------------------
template<int K, bool GELU>
__device__ __forceinline__ void gemm_tile_f32(
    const float* __restrict__ Alds,
    const float* __restrict__ W,
    const float* __restrict__ bias,
    const float* __restrict__ Res,      // optional residual, stride 256
    float* __restrict__ Out, int outStride,
    int n0, int lane) {
  const int half = lane >> 4;           // 0: K=k0,k0+1  1: K=k0+2,k0+3
  const int ml   = lane & 15;
  v8f acc = {0.f, 0.f, 0.f, 0.f, 0.f, 0.f, 0.f, 0.f};
  const float* arow = Alds + ml * K + 2 * half;                 // A[ml][*]
  const float* wrow = W + (size_t)(n0 + ml) * K + 2 * half;     // W[n0+ml][*]
  for (int k0 = 0; k0 < K; k0 += 4) {
    v2f a, b;
    a[0] = arow[k0];  a[1] = arow[k0 + 1];
    b[0] = wrow[k0];  b[1] = wrow[k0 + 1];
    acc = __builtin_amdgcn_wmma_f32_16x16x4_f32(
        false, a, false, b, (short)0, acc, false, false);
  }
  const float bb = bias ? bias[n0 + ml] : 0.0f;
  for (int r = 0; r < 8; ++r) {
    const int m = r + 8 * half;         // D layout: vgpr r -> rows r / r+8
    float v = acc[r] + bb;
    if (Res)  v += Res[m * 256 + (n0 + ml)];
    if (GELU) v = gelu_exact(v);
    Out[m * outStride + (n0 + ml)] = v;
  }
}

// ---------------------------------------------------------------------------
// Kernel 2: the entire token-mixing chain on (4,256), single workgroup.
// 512 threads = 16 waves; one wave per 16-wide N tile in each GEMM;
// one wave per row for the LN/GN/gelu stages.
// ---------------------------------------------------------------------------
__global__ __launch_bounds__(512) void token_mix_kernel(
    const float* __restrict__ tokens,      // [4][256]
    const float* __restrict__ in_proj_w,   // [768][256]
    const float* __restrict__ in_proj_b,   // [768]
    const float* __restrict__ out_proj_w,  // [256][256]
    const float* __restrict__ out_proj_b,  // [256]
    const float* __restrict__ ln1_w, const float* __restrict__ ln1_b,
    const float* __restrict__ ln2_w, const float* __restrict__ ln2_b,
    const float* __restrict__ dw_w,  const float* __restrict__ dw_b,   // [256][1][3],[256]
    const float* __restrict__ gn1_w, const float* __restrict__ gn1_b,
    const float* __restrict__ pw_w,  const float* __restrict__ pw_b,   // [256][256][1],[256]
    const float* __restrict__ gn2_w, const float* __restrict__ gn2_b,
    const float* __restrict__ ffn_w1, const float* __restrict__ ffn_b1,// [128][256],[128]
    const float* __restrict__ ffn_w2, const float* __restrict__ ffn_b2,// [256][128],[256]
    float* __restrict__ tok2) {            // [4][256] out
  __shared__ float sA[16 * 256];
  __shared__ float sB[16 * 256];
  __shared__ float sC[16 * 256];
  __shared__ float sH[16 * 128];

  const int tid  = threadIdx.x;
  const int lane = tid & 31;
  const int wave = tid >> 5;
  const int n0   = wave * 16;

  // Stage A: load tokens, pad rows 4..15 with zero
  for (int i = tid; i < 16 * 256; i += 512) {
    sA[i] = ((i >> 8) < B_DIM) ? tokens[i] : 0.0f;
  }
  __syncthreads();

  // GEMM1: v = tokens @ wv^T + bv    (wv = in_proj_w[2C:], bv = in_proj_b[2C:])
  gemm_tile_f32<256, false>(sA, in_proj_w + 2 * C_DIM * C_DIM,
                            in_proj_b + 2 * C_DIM, nullptr, sB, 256, n0, lane);
  __syncthreads();

  // GEMM2: sC = v @ out_proj_w^T + out_proj_b + tokens   (residual fused)
  gemm_tile_f32<256, false>(sB, out_proj_w, out_proj_b, sA, sC, 256, n0, lane);
  __syncthreads();

  // LN1: sA[r] = LN(sC[r]) * ln1_w + ln1_b
  if (wave < B_DIM) {
    const float* row = sC + wave * 256;
    float s = 0.f, s2 = 0.f;
    for (int i = lane; i < 256; i += 32) { float v = row[i]; s += v; s2 += v * v; }
    wave_reduce2(s, s2);
    const float mu = s * (1.f / 256.f);
    const float rs = rsqrtf(s2 * (1.f / 256.f) - mu * mu + EPS_LN);
    for (int i = lane; i < 256; i += 32)
      sA[wave * 256 + i] = (row[i] - mu) * rs * ln1_w[i] + ln1_b[i];
  }
  __syncthreads();

  // depthwise scale + GN1 + gelu: sB[r] = gelu(GN1(tok * dw_w[:,0,1] + dw_b))
  if (wave < B_DIM) {
    float cv[8];
    float s = 0.f, s2 = 0.f;
    for (int j = 0; j < 8; ++j) {
      const int i = lane + 32 * j;
      const float v = sA[wave * 256 + i] * dw_w[i * 3 + 1] + dw_b[i];
      cv[j] = v; s += v; s2 += v * v;
    }
    wave_reduce2(s, s2);
    const float mu = s * (1.f / 256.f);
    const float rs = rsqrtf(s2 * (1.f / 256.f) - mu * mu + EPS_LN);
    for (int j = 0; j < 8; ++j) {
      const int i = lane + 32 * j;
      sB[wave * 256 + i] = gelu_exact((cv[j] - mu) * rs * gn1_w[i] + gn1_b[i]);
    }
  }
  __syncthreads();

  // GEMM3: sC = c @ pw_w[:,:,0]^T + pw_b
  gemm_tile_f32<256, false>(sB, pw_w, pw_b, nullptr, sC, 256, n0, lane);
  __syncthreads();

  // GN2, residual tok += c, then LN2 into sB
  if (wave < B_DIM) {
    const float* row = sC + wave * 256;
    float s = 0.f, s2 = 0.f;
    for (int i = lane; i < 256; i += 32) { float v = row[i]; s += v; s2 += v * v; }
    wave_reduce2(s, s2);
    const float mu = s * (1.f / 256.f);
    const float rs = rsqrtf(s2 * (1.f / 256.f) - mu * mu + EPS_LN);
    float tn[8]; float ts = 0.f, ts2 = 0.f;
    for (int j = 0; j < 8; ++j) {
      const int i = lane + 32 * j;
      const float g = (row[i] - mu) * rs * gn2_w[i] + gn2_b[i];
      const float t = sA[wave * 256 + i] + g;
      tn[j] = t; ts += t; ts2 += t * t;
      sA[wave * 256 + i] = t;                       // tok (for final residual)
    }
    wave_reduce2(ts, ts2);
    const float mu2 = ts * (1.f / 256.f);
    const float rs2 = rsqrtf(ts2 * (1.f / 256.f) - mu2 * mu2 + EPS_LN);
    for (int j = 0; j < 8; ++j) {
      const int i = lane + 32 * j;
      sB[wave * 256 + i] = (tn[j] - mu2) * rs2 * ln2_w[i] + ln2_b[i];
    }
  }
  __syncthreads();

  // GEMM4: sH = gelu(t2 @ ffn_w1^T + ffn_b1)   (N=128 -> waves 0..7)
  if (wave < 8)
    gemm_tile_f32<256, true>(sB, ffn_w1, ffn_b1, nullptr, sH, 128, n0, lane);
  __syncthreads();

  // GEMM5: sC = h @ ffn_w2^T + ffn_b2          (K=128, N=256)
  gemm_tile_f32<128, false>(sH, ffn_w2, ffn_b2, nullptr, sC, 256, n0, lane);
  __syncthreads();

  // tok2 = tok + f  (rows 0..3)
  for (int i = tid; i < B_DIM * 256; i += 512)
    tok2[i] = sA[i] + sC[i];
}

// ---------------------------------------------------------------------------
// Kernel 3: out = x * tok2[b,c]  (streaming, b128 in / b128 out)
// ---------------------------------------------------------------------------
__global__ __launch_bounds__(256) void scale_kernel(
    const float* __restrict__ x, const float* __restrict__ tok2,
    float* __restrict__ out) {
  const size_t i = (size_t)blockIdx.x * 256 + threadIdx.x;  // float4 index
  const float4 v = ((const float4*)x)[i];
  const float sc = tok2[i >> 14];                           // 16384 float4 per (b,c)
  float4 r;
  r.x = v.x * sc; r.y = v.y * sc; r.z = v.z * sc; r.w = v.w * sc;
  ((float4*)out)[i] = r;
}

extern "C" void kernel_launch(void* const* d_in, const int* in_sizes, int n_in,
                              void* d_out, int out_size, void* d_ws, size_t ws_size,
                              hipStream_t stream) {
  const float* x          = (const float*)d_in[0];
  const float* in_proj_w  = (const float*)d_in[1];
  const float* in_proj_b  = (const float*)d_in[2];
  const float* out_proj_w = (const float*)d_in[3];
  const float* out_proj_b = (const float*)d_in[4];
  const float* ln1_w      = (const float*)d_in[5];
  const float* ln1_b      = (const float*)d_in[6];
  const float* ln2_w      = (const float*)d_in[7];
  const float* ln2_b      = (const float*)d_in[8];
  const float* dw_w       = (const float*)d_in[9];
  const float* dw_b       = (const float*)d_in[10];
  const float* gn1_w      = (const float*)d_in[11];
  const float* gn1_b      = (const float*)d_in[12];
  const float* pw_w       = (const float*)d_in[13];
  const float* pw_b       = (const float*)d_in[14];
  const float* gn2_w      = (const float*)d_in[15];
  const float* gn2_b      = (const float*)d_in[16];
  const float* ffn_w1     = (const float*)d_in[17];
  const float* ffn_b1     = (const float*)d_in[18];
  const float* ffn_w2     = (const float*)d_in[19];
  const float* ffn_b2     = (const float*)d_in[20];

  float* tokens = (float*)d_ws;            // 1024 floats
  float* tok2   = tokens + 1024;           // 1024 floats

  // 1) spatial mean: one block per (b,c)
  tokens_mean_kernel<<<B_DIM * C_DIM, 256, 0, stream>>>(x, tokens);

  // 2) full token-mixing chain in one workgroup (fp32 WMMA GEMMs)
  token_mix_kernel<<<1, 512, 0, stream>>>(
      tokens, in_proj_w, in_proj_b, out_proj_w, out_proj_b,
      ln1_w, ln1_b, ln2_w, ln2_b, dw_w, dw_b, gn1_w, gn1_b,
      pw_w, pw_b, gn2_w, gn2_b, ffn_w1, ffn_b1, ffn_w2, ffn_b2, tok2);

  // 3) broadcast scale: 16,777,216 float4 elements
  const int total4 = (B_DIM * C_DIM * SPATIAL) / 4;
  scale_kernel<<<total4 / 256, 256, 0, stream>>>(x, tok2, (float*)d_out);
}